// IntervalClusterTripletFT_89146341196572
// MI455X (gfx1250) — compile-verified
//
#include <hip/hip_runtime.h>

// CDNA5 / gfx1250 fused triplet-margin loss with hard mining.
// dist^2(i,j) = x2[i] + x2[j] - 2*G[i,j], G = X X^T via V_WMMA_F32_16X16X4_F32.
// Per row keep: posMax = max over same-cluster dist^2, negMin = min over
// cross-cluster dist^2. loss = mean(max(sqrt(posMax) - sqrt(negMin) + 1, 0)).
//
// v2: 4 WMMA chains per wave (panel width 128, K-chunk 32) + register-staged
// global prefetch of the next B chunk to overlap VMEM with the WMMA stream.

typedef __attribute__((ext_vector_type(2))) float v2f;
typedef __attribute__((ext_vector_type(8))) float v8f;

#define EMB      128
#define N_TOT    8192          // 1024 clusters * 8
#define MARGIN   1.0f

#define MBLK     64            // rows per workgroup
#define NPAN     128           // column panel width
#define KCHUNK   32            // K chunk resident for B
#define A_STRIDE 132           // 128 + 4 pad -> conflict-free v2f reads
#define B_STRIDE 36            // 32 + 4 pad  -> conflict-free v2f reads
#define NITER    ((N_TOT / NPAN) * (EMB / KCHUNK))   // 64 * 4 = 256

// ---------------- kernel 1: row squared norms + zero the accumulator -------
__global__ __launch_bounds__(256)
void rownorm_kernel(const float* __restrict__ x, float* __restrict__ x2,
                    float* __restrict__ accum) {
  const int row  = blockIdx.x * 8 + (threadIdx.x >> 5);  // one wave32 per row
  const int lane = threadIdx.x & 31;
  const float4 v = *(const float4*)(x + (size_t)row * EMB + lane * 4);
  float s = v.x * v.x + v.y * v.y + v.z * v.z + v.w * v.w;
  #pragma unroll
  for (int m = 16; m >= 1; m >>= 1) s += __shfl_xor(s, m, 32);
  if (lane == 0) x2[row] = s;
  if (blockIdx.x == 0 && threadIdx.x == 0) *accum = 0.0f;  // self-initializing
}

// ---------------- kernel 2: fused Gram GEMM + masked row min/max -----------
__global__ __launch_bounds__(256)
void triplet_gemm_kernel(const float* __restrict__ x,
                         const float* __restrict__ x2,
                         float* __restrict__ accum) {
  __shared__ float smem[MBLK * A_STRIDE + NPAN * B_STRIDE];  // 52,224 bytes
  float* As = smem;                       // [64][132] full-K A block
  float* Bs = smem + MBLK * A_STRIDE;     // [128][36] K-chunked B panel

  const int tid   = threadIdx.x;
  const int lane  = tid & 31;
  const int wave  = tid >> 5;        // 0..7
  const int hi    = lane >> 4;       // WMMA K-half / M-half selector
  const int ln    = lane & 15;       // M (A) / N (B) within a 16-tile
  const int mtile = wave >> 1;       // 0..3
  const int ntBase = (wave & 1) * 4; // this wave's 4 N-tiles in the panel
  const int rowBase = blockIdx.x * MBLK;

  // Load the 64x128 A block (full K) into LDS, float4-coalesced.
  #pragma unroll
  for (int q = 0; q < 8; ++q) {
    const int idx = tid + 256 * q;
    const int r = idx >> 5, c4 = idx & 31;
    *(float4*)(As + r * A_STRIDE + c4 * 4) =
        *(const float4*)(x + (size_t)(rowBase + r) * EMB + c4 * 4);
  }

  // Rows owned by this thread's accumulators: i = iBase + v, v=0..7.
  // iBase is 8-aligned -> all 8 rows share one cluster id.
  const int iBase = rowBase + mtile * 16 + 8 * hi;
  const int ci    = iBase >> 3;
  float x2i[8];
  #pragma unroll
  for (int v = 0; v < 8; ++v) x2i[v] = x2[iBase + v];

  float negMin[8], posMax[8];
  #pragma unroll
  for (int v = 0; v < 8; ++v) { negMin[v] = __builtin_inff(); posMax[v] = 0.0f; }

  // B-panel cooperative fill addressing: 128 rows x 32 floats per chunk,
  // 4 float4 per thread (rows bRow + 32q).
  const int bRow = tid >> 3;          // 0..31
  const int bCol = (tid & 7) * 4;     // 0..28
  const int bst  = bRow * B_STRIDE + bCol;

  // Prefetch chunk 0 (j0 = 0, kk = 0) into registers.
  float4 pre[4];
  {
    const float* gp = x + (size_t)bRow * EMB + bCol;
    #pragma unroll
    for (int q = 0; q < 4; ++q)
      pre[q] = *(const float4*)(gp + (size_t)q * 32 * EMB);
  }

  // Per-wave B read pointers (chunk-invariant).
  const float* pB0 = Bs + ((ntBase + 0) * 16 + ln) * B_STRIDE + 2 * hi;
  const float* pB1 = Bs + ((ntBase + 1) * 16 + ln) * B_STRIDE + 2 * hi;
  const float* pB2 = Bs + ((ntBase + 2) * 16 + ln) * B_STRIDE + 2 * hi;
  const float* pB3 = Bs + ((ntBase + 3) * 16 + ln) * B_STRIDE + 2 * hi;

  v8f acc0, acc1, acc2, acc3;

  for (int it = 0; it < NITER; ++it) {
    const int j0 = (it >> 2) * NPAN;
    const int kk = (it & 3) * KCHUNK;

    __syncthreads();  // previous chunk's LDS reads complete
    #pragma unroll
    for (int q = 0; q < 4; ++q)
      *(float4*)(Bs + bst + q * 32 * B_STRIDE) = pre[q];
    __syncthreads();  // Bs (and initial As) visible

    // Prefetch the NEXT chunk while this chunk's WMMAs run.
    if (it + 1 < NITER) {
      const int jn = ((it + 1) >> 2) * NPAN;
      const int kn = ((it + 1) & 3) * KCHUNK;
      const float* gp = x + (size_t)(jn + bRow) * EMB + kn + bCol;
      #pragma unroll
      for (int q = 0; q < 4; ++q)
        pre[q] = *(const float4*)(gp + (size_t)q * 32 * EMB);
    }

    if ((it & 3) == 0) {
      #pragma unroll
      for (int v = 0; v < 8; ++v) {
        acc0[v] = 0.0f; acc1[v] = 0.0f; acc2[v] = 0.0f; acc3[v] = 0.0f;
      }
    }

    const float* pA = As + (mtile * 16 + ln) * A_STRIDE + kk + 2 * hi;
    #pragma unroll
    for (int k = 0; k < KCHUNK; k += 4) {
      const v2f a  = *(const v2f*)(pA  + k);
      const v2f b0 = *(const v2f*)(pB0 + k);
      const v2f b1 = *(const v2f*)(pB1 + k);
      const v2f b2 = *(const v2f*)(pB2 + k);
      const v2f b3 = *(const v2f*)(pB3 + k);
      acc0 = __builtin_amdgcn_wmma_f32_16x16x4_f32(false, a, false, b0,
                                                   (short)0, acc0, false, false);
      acc1 = __builtin_amdgcn_wmma_f32_16x16x4_f32(false, a, false, b1,
                                                   (short)0, acc1, false, false);
      acc2 = __builtin_amdgcn_wmma_f32_16x16x4_f32(false, a, false, b2,
                                                   (short)0, acc2, false, false);
      acc3 = __builtin_amdgcn_wmma_f32_16x16x4_f32(false, a, false, b3,
                                                   (short)0, acc3, false, false);
    }

    if ((it & 3) == 3) {  // K complete for this panel: masked min/max update
      const int jn0 = j0 + (ntBase + 0) * 16 + ln;
      const int jn1 = j0 + (ntBase + 1) * 16 + ln;
      const int jn2 = j0 + (ntBase + 2) * 16 + ln;
      const int jn3 = j0 + (ntBase + 3) * 16 + ln;
      const float x2j0 = x2[jn0], x2j1 = x2[jn1];
      const float x2j2 = x2[jn2], x2j3 = x2[jn3];
      const bool s0 = (ci == (jn0 >> 3));
      const bool s1 = (ci == (jn1 >> 3));
      const bool s2 = (ci == (jn2 >> 3));
      const bool s3 = (ci == (jn3 >> 3));
      #pragma unroll
      for (int v = 0; v < 8; ++v) {
        const float sq0 = fmaxf(fmaf(-2.0f, acc0[v], x2i[v] + x2j0), 0.0f);
        const float sq1 = fmaxf(fmaf(-2.0f, acc1[v], x2i[v] + x2j1), 0.0f);
        const float sq2 = fmaxf(fmaf(-2.0f, acc2[v], x2i[v] + x2j2), 0.0f);
        const float sq3 = fmaxf(fmaf(-2.0f, acc3[v], x2i[v] + x2j3), 0.0f);
        if (s0) posMax[v] = fmaxf(posMax[v], sq0);
        else    negMin[v] = fminf(negMin[v], sq0);
        if (s1) posMax[v] = fmaxf(posMax[v], sq1);
        else    negMin[v] = fminf(negMin[v], sq1);
        if (s2) posMax[v] = fmaxf(posMax[v], sq2);
        else    negMin[v] = fminf(negMin[v], sq2);
        if (s3) posMax[v] = fmaxf(posMax[v], sq3);
        else    negMin[v] = fminf(negMin[v], sq3);
      }
    }
  }

  // Cross-lane / cross-wave reduction (reuse LDS; GEMM phase is done).
  __syncthreads();
  float* redN = smem;          // 2048 floats
  float* redP = smem + 2048;   // 2048 floats
  #pragma unroll
  for (int v = 0; v < 8; ++v) {
    redN[tid * 8 + v] = negMin[v];
    redP[tid * 8 + v] = posMax[v];
  }
  __syncthreads();

  if (tid < MBLK) {
    // row i_local = tid  ->  mtile = tid/16, hi = (tid>>3)&1, v = tid&7
    const int mt = tid >> 4, h = (tid >> 3) & 1, v = tid & 7;
    float nm = __builtin_inff(), pm = 0.0f;
    #pragma unroll
    for (int w2 = 0; w2 < 2; ++w2) {
      const int wv = mt * 2 + w2;
      for (int l = 0; l < 16; ++l) {
        const int t = wv * 32 + h * 16 + l;
        nm = fminf(nm, redN[t * 8 + v]);
        pm = fmaxf(pm, redP[t * 8 + v]);
      }
    }
    const float an    = sqrtf(nm);
    const float apd   = sqrtf(pm);
    const float hinge = fmaxf(apd - an + MARGIN, 0.0f);
    atomicAdd(accum, hinge);
  }
}

// ---------------- kernel 3: finalize mean ---------------------------------
__global__ void finalize_kernel(const float* __restrict__ accum,
                                float* __restrict__ out) {
  if (threadIdx.x == 0 && blockIdx.x == 0)
    out[0] = accum[0] * (1.0f / (float)N_TOT);
}

extern "C" void kernel_launch(void* const* d_in, const int* in_sizes, int n_in,
                              void* d_out, int out_size, void* d_ws, size_t ws_size,
                              hipStream_t stream) {
  const float* x   = (const float*)d_in[0];   // [1024,8,128] fp32 -> [8192,128]
  float*       out = (float*)d_out;           // scalar fp32
  float*       ws  = (float*)d_ws;
  float* accum = ws;                          // ws[0]
  float* x2    = ws + 64;                     // 8192 floats, 256B-offset

  rownorm_kernel<<<N_TOT / 8, 256, 0, stream>>>(x, x2, accum);
  triplet_gemm_kernel<<<N_TOT / MBLK, 256, 0, stream>>>(x, x2, accum);
  finalize_kernel<<<1, 1, 0, stream>>>(accum, out);
}